// compDiffDist_batch_14499809591883
// MI455X (gfx1250) — compile-verified
//
#include <hip/hip_runtime.h>
#include <math.h>

typedef __attribute__((ext_vector_type(2))) float v2f;
typedef __attribute__((ext_vector_type(8))) float v8f;

#define D 64   // feature dimension (fixed by reference)

// ---------------------------------------------------------------------------
// Kernel 1: w[b,k] = max(G[b,k],0)^(2t);  s[b,i] = sum_k w[b,k] * V[b,i,k]^2
// ---------------------------------------------------------------------------
__global__ __launch_bounds__(256) void prep_kernel(
    const float* __restrict__ G, const float* __restrict__ V,
    const float* __restrict__ t_ptr, float* __restrict__ w_out,
    float* __restrict__ s_out, int r)
{
    __shared__ float wlds[D];
    const int b   = blockIdx.y;
    const int tid = threadIdx.x;
    const float t2 = 2.0f * t_ptr[0];
    if (tid < D) {
        float wv = powf(fmaxf(G[b * D + tid], 0.0f), t2);
        wlds[tid] = wv;
        if (blockIdx.x == 0) w_out[b * D + tid] = wv;
    }
    __syncthreads();

    const int i = blockIdx.x * 256 + tid;
    const float* vrow = V + ((size_t)b * r + i) * D;
    float acc = 0.0f;
#pragma unroll
    for (int k = 0; k < D; k += 4) {
        float4 v4 = *(const float4*)(vrow + k);
        acc += wlds[k + 0] * v4.x * v4.x;
        acc += wlds[k + 1] * v4.y * v4.y;
        acc += wlds[k + 2] * v4.z * v4.z;
        acc += wlds[k + 3] * v4.w * v4.w;
    }
    s_out[b * r + i] = acc;
}

// ---------------------------------------------------------------------------
// Kernel 2: y[b,i,j] = s[i] + s[j] - 2 * sum_k w[k] V[b,i,k] V[b,j,k]
// 4 waves / block; each wave computes a 64x64 tile as a 4x4 grid of
// V_WMMA_F32_16X16X4_F32 tiles, K=64 accumulated in 16 chunks of 4.
// ---------------------------------------------------------------------------
__global__ __launch_bounds__(128) void dist_wmma_kernel(
    const float* __restrict__ V, const float* __restrict__ w,
    const float* __restrict__ s, float* __restrict__ out, int r)
{
    __shared__ float wlds[D];
    const int b   = blockIdx.z;
    const int tid = threadIdx.x;
    if (tid < D) wlds[tid] = w[b * D + tid];
    __syncthreads();

    const int wave = tid >> 5;
    const int lane = tid & 31;
    const int half = lane >> 4;   // lanes 16-31 carry K+2,K+3 of each chunk
    const int l16  = lane & 15;   // A: M row / B: N col within 16-wide tile

    const int wi = blockIdx.y * 128 + (wave >> 1) * 64;  // wave's row base
    const int wj = blockIdx.x * 128 + (wave & 1) * 64;   // wave's col base

    // Per-lane weights for each K-chunk: float2 at wlds[4*kc + 2*half]
    v2f wv[16];
#pragma unroll
    for (int kc = 0; kc < 16; ++kc)
        wv[kc] = *(const v2f*)(&wlds[kc * 4 + 2 * half]);

    const float* baseA[4];
    const float* baseB[4];
#pragma unroll
    for (int tI = 0; tI < 4; ++tI)
        baseA[tI] = V + ((size_t)b * r + (wi + tI * 16 + l16)) * D + 2 * half;
#pragma unroll
    for (int tJ = 0; tJ < 4; ++tJ)
        baseB[tJ] = V + ((size_t)b * r + (wj + tJ * 16 + l16)) * D + 2 * half;

    v8f acc[4][4] = {};

#pragma unroll
    for (int kc = 0; kc < 16; ++kc) {
        v2f a[4], bb[4];
#pragma unroll
        for (int tI = 0; tI < 4; ++tI)
            a[tI] = *(const v2f*)(baseA[tI] + kc * 4) * wv[kc];   // A = w ⊙ V
#pragma unroll
        for (int tJ = 0; tJ < 4; ++tJ)
            bb[tJ] = *(const v2f*)(baseB[tJ] + kc * 4);           // B = Vᵀ tile
#pragma unroll
        for (int tI = 0; tI < 4; ++tI)
#pragma unroll
            for (int tJ = 0; tJ < 4; ++tJ)
                acc[tI][tJ] = __builtin_amdgcn_wmma_f32_16x16x4_f32(
                    /*neg_a=*/false, a[tI], /*neg_b=*/false, bb[tJ],
                    /*c_mod=*/(short)0, acc[tI][tJ],
                    /*reuse_a=*/false, /*reuse_b=*/false);
    }

    // Epilogue: y = s_i + s_j - 2*cross, fused into the store.
    const float* sb = s + (size_t)b * r;
    float scol[4];
#pragma unroll
    for (int tJ = 0; tJ < 4; ++tJ)
        scol[tJ] = sb[wj + tJ * 16 + l16];

#pragma unroll
    for (int tI = 0; tI < 4; ++tI) {
        float srow[8];
#pragma unroll
        for (int v = 0; v < 8; ++v)
            srow[v] = sb[wi + tI * 16 + 8 * half + v];
#pragma unroll
        for (int tJ = 0; tJ < 4; ++tJ) {
            const int col = wj + tJ * 16 + l16;
#pragma unroll
            for (int v = 0; v < 8; ++v) {
                const int row = wi + tI * 16 + 8 * half + v;
                out[((size_t)b * r + row) * r + col] =
                    fmaf(-2.0f, acc[tI][tJ][v], srow[v] + scol[tJ]);
            }
        }
    }
}

// ---------------------------------------------------------------------------
// Launch: d_in[0]=G (bs*64), d_in[1]=V (bs*r*64), d_in[2]=t (1)
// d_ws: w (bs*64 floats) followed by s (bs*r floats)
// ---------------------------------------------------------------------------
extern "C" void kernel_launch(void* const* d_in, const int* in_sizes, int n_in,
                              void* d_out, int out_size, void* d_ws, size_t ws_size,
                              hipStream_t stream) {
    const float* G = (const float*)d_in[0];
    const float* V = (const float*)d_in[1];
    const float* t = (const float*)d_in[2];
    float* out = (float*)d_out;

    const int bs = in_sizes[0] / D;              // 4
    const int r  = in_sizes[1] / (bs * D);       // 4096

    float* w = (float*)d_ws;                     // bs*64 floats
    float* s = w + (size_t)bs * D;               // bs*r floats

    dim3 g1(r / 256, bs);
    prep_kernel<<<g1, 256, 0, stream>>>(G, V, t, w, s, r);

    dim3 g2(r / 128, r / 128, bs);
    dist_wmma_kernel<<<g2, 128, 0, stream>>>(V, w, s, out, r);
}